// Spatiotemporal_704374636831
// MI455X (gfx1250) — compile-verified
//
#include <hip/hip_runtime.h>
#include <hip/hip_bf16.h>

typedef __attribute__((ext_vector_type(2))) float v2f;
typedef __attribute__((ext_vector_type(8))) float v8f;

#define LOG2PIF 1.8378770664093453f
#define LOG2EF  1.4426950408889634f
#define NCAND 625
#define NCPAD 640      /* 40 tiles of 16 */
#define HMAX 400
#define HPAD 416

// ---------------------------------------------------------------------------
// Kernel 1: grid bounds (from history points 0 and 1 only — reference quirk),
// candidate grid + squared norms. One small block.
// ---------------------------------------------------------------------------
__global__ void setup_grid_kernel(const float* __restrict__ hist,
                                  float* __restrict__ gx, float* __restrict__ gy,
                                  float* __restrict__ gn) {
    __shared__ float r0[64], r1[64], r2[64], r3[64];
    const int tid = threadIdx.x;  // 64 threads, one per batch
    float x0 = hist[(tid * 384 + 0) * 2 + 0];
    float x1 = hist[(tid * 384 + 0) * 2 + 1];
    float y0 = hist[(tid * 384 + 1) * 2 + 0];
    float y1 = hist[(tid * 384 + 1) * 2 + 1];
    r0[tid] = fminf(x0, x1); r1[tid] = fmaxf(x0, x1);
    r2[tid] = fminf(y0, y1); r3[tid] = fmaxf(y0, y1);
    __syncthreads();
    for (int s = 32; s > 0; s >>= 1) {
        if (tid < s) {
            r0[tid] = fminf(r0[tid], r0[tid + s]);
            r1[tid] = fmaxf(r1[tid], r1[tid + s]);
            r2[tid] = fminf(r2[tid], r2[tid + s]);
            r3[tid] = fmaxf(r3[tid], r3[tid + s]);
        }
        __syncthreads();
    }
    const float minx = r0[0], maxx = r1[0], miny = r2[0], maxy = r3[0];
    const float dx = (maxx - minx) / 24.0f;
    const float dy = (maxy - miny) / 24.0f;
    for (int c = tid; c < NCPAD; c += 64) {
        float x = 0.f, y = 0.f;
        if (c < NCAND) {
            int jj = c % 25, ii = c / 25;   // arr[k] = (xlim[j], ylim[i]), k=i*25+j
            x = minx + dx * (float)jj;
            y = miny + dy * (float)ii;
        }
        gx[c] = x; gy[c] = y; gn[c] = x * x + y * y;
    }
}

// ---------------------------------------------------------------------------
// Kernel 2: init growing history (SoA in workspace) from inputs.
// ---------------------------------------------------------------------------
__global__ void init_hist_kernel(const float* __restrict__ hist,
                                 const float* __restrict__ itime,
                                 float* __restrict__ hx, float* __restrict__ hy,
                                 float* __restrict__ tt) {
    int idx = blockIdx.x * 256 + threadIdx.x;  // over 64*384
    if (idx < 64 * 384) {
        int b = idx / 384, j = idx % 384;
        hx[b * HMAX + j] = hist[idx * 2 + 0];
        hy[b * HMAX + j] = hist[idx * 2 + 1];
        tt[b * HMAX + j] = itime[idx];
    }
}

// ---------------------------------------------------------------------------
// Kernel 3: one prediction step. Block = batch (256 thr = 8 wave32).
// S_c = sum_j exp(logw_j - 0.5*sq_cj/sigma^2), argmin_c (first-index ties).
// V_WMMA_F32_16X16X4_F32 produces LOG2E * exponent directly (the padded K=2,3
// slots carry the biases, and log2e is folded into all A rows):
//   A[m, 0:2] = log2e/s^2 * cand_m ; A[m,2] = log2e ; A[m,3] = -log2e*|c_m|^2/(2s^2)
//   B[0:2, n] = hist_n             ; B[2,n] = logw_n - |h_n|^2/(2s^2) ; B[3,n] = 1
//   D[m][n]   = log2e * (logw_n - |c_m - h_n|^2/(2s^2)),  C = inline 0
// so each output element needs exactly one bare v_exp_f32 (exp2) + one add.
// Both half-waves read their A/B rows from dedicated LDS tables -> uniform
// ds_load_b64; B is software-prefetched one tile ahead.
// ---------------------------------------------------------------------------
__global__ __launch_bounds__(256) void step_kernel(
    const float* __restrict__ curr_time,
    const float* __restrict__ beta_p, const float* __restrict__ log_sigma_p,
    const float* __restrict__ gx, const float* __restrict__ gy,
    const float* __restrict__ gn,
    float* __restrict__ hx_g, float* __restrict__ hy_g, float* __restrict__ t_g,
    float* __restrict__ out, int step)
{
    __shared__ float2 sBM[2 * HPAD];    // [j]=(hx,hy)  [HPAD+j]=(basen,1)
    __shared__ float2 sAM[2 * NCPAD];   // [c]=le*(sA*cx,sA*cy) [NCPAD+c]=le*(1,-inv2s2*cn)
    __shared__ float  sLw[HPAD];
    __shared__ float  sS[NCPAD];
    __shared__ float  sRedV[256];
    __shared__ int    sRedI[256];

    const int b = blockIdx.x;
    const int tid = threadIdx.x;
    const int L = 384 + step;
    const int nt = (L + 15) >> 4;
    const float beta = beta_p[0];
    const float sigma = __expf(log_sigma_p[0]);
    const float inv2s2 = 0.5f / (sigma * sigma);
    const float sA = 2.0f * inv2s2;
    const float t = curr_time[b * 16 + step];

    // Stage history coords + raw log-weights; build A tables (log2e folded in).
    for (int j = tid; j < HPAD; j += 256) {
        float x = 0.f, y = 0.f, lw = -1e30f;
        if (j < L) {
            x = hx_g[b * HMAX + j];
            y = hy_g[b * HMAX + j];
            lw = -beta * (t - t_g[b * HMAX + j]);
        }
        sBM[j] = make_float2(x, y);
        sLw[j] = lw;
    }
    for (int c = tid; c < NCPAD; c += 256) {
        float x = gx[c], y = gy[c], n = gn[c];
        sAM[c] = make_float2(LOG2EF * sA * x, LOG2EF * sA * y);
        sAM[NCPAD + c] = make_float2(LOG2EF, -LOG2EF * inv2s2 * n);
        sS[c] = __builtin_inff();
    }
    __syncthreads();

    // Block softmax-normalization constant for the log-weights.
    float m = -1e30f;
    for (int j = tid; j < L; j += 256) m = fmaxf(m, sLw[j]);
    sRedV[tid] = m; __syncthreads();
    for (int s = 128; s > 0; s >>= 1) {
        if (tid < s) sRedV[tid] = fmaxf(sRedV[tid], sRedV[tid + s]);
        __syncthreads();
    }
    m = sRedV[0]; __syncthreads();
    float sum = 0.f;
    for (int j = tid; j < L; j += 256) sum += __expf(sLw[j] - m);
    sRedV[tid] = sum; __syncthreads();
    for (int s = 128; s > 0; s >>= 1) {
        if (tid < s) sRedV[tid] += sRedV[tid + s];
        __syncthreads();
    }
    const float lse = m + __logf(sRedV[0]);
    __syncthreads();
    // Second B table: K=2,3 rows = (basen_j, 1).
    for (int j = tid; j < HPAD; j += 256) {
        float2 h = sBM[j];
        float lw = (j < L) ? (sLw[j] - lse) : -1e30f;
        sBM[HPAD + j] = make_float2(lw - inv2s2 * (h.x * h.x + h.y * h.y), 1.0f);
    }
    __syncthreads();

    // WMMA phase: wave w owns candidate tiles {w, w+8, ..., w+32}.
    const int lane = tid & 31;
    const int wv = tid >> 5;
    const int half = lane >> 4;     // 0 -> K=0,1 rows ; 1 -> K=2,3 rows
    const int l16 = lane & 15;
    const int aoff = half * NCPAD + l16;
    const int boff = half * HPAD + l16;

    v2f a[5];
    float acc[5][8];
    #pragma unroll
    for (int k = 0; k < 5; ++k) {
        const int ct = wv + 8 * k;
        float2 av = sAM[aoff + ct * 16];
        a[k].x = av.x; a[k].y = av.y;
        #pragma unroll
        for (int v = 0; v < 8; ++v) acc[k][v] = 0.f;
    }

    float2 bv = sBM[boff];                       // prime the pipeline
    for (int ht = 0; ht < nt; ++ht) {
        const int nx = (ht + 1 < nt) ? (ht + 1) : ht;
        float2 bnext = sBM[boff + nx * 16];      // prefetch next tile
        v2f bm; bm.x = bv.x; bm.y = bv.y;
        const v8f z = {};
        #pragma unroll
        for (int k = 0; k < 5; ++k) {
            v8f d = __builtin_amdgcn_wmma_f32_16x16x4_f32(
                false, a[k], false, bm, (short)0, z, false, false);
            float e[8];
            #pragma unroll
            for (int v = 0; v < 8; ++v) e[v] = __builtin_amdgcn_exp2f(d[v]);
            #pragma unroll
            for (int v = 0; v < 8; ++v) acc[k][v] += e[v];
        }
        bv = bnext;
    }

    // Reduce over the 16 lanes (n dim) of each half; store per-candidate sums.
    #pragma unroll
    for (int k = 0; k < 5; ++k) {
        const int ct = wv + 8 * k;
        #pragma unroll
        for (int v = 0; v < 8; ++v) {
            float x = acc[k][v];
            x += __shfl_xor(x, 1, 32);
            x += __shfl_xor(x, 2, 32);
            x += __shfl_xor(x, 4, 32);
            x += __shfl_xor(x, 8, 32);
            acc[k][v] = x;
        }
        if (l16 == 0) {
            #pragma unroll
            for (int v = 0; v < 8; ++v) {
                int c = ct * 16 + half * 8 + v;
                if (c < NCAND) sS[c] = acc[k][v];
            }
        }
    }
    __syncthreads();

    // Block argmin with first-index tie-breaking (matches jnp.argmin).
    float bvv = __builtin_inff(); int bii = 0;
    for (int c = tid; c < NCPAD; c += 256) {
        float v = sS[c];
        if (v < bvv) { bvv = v; bii = c; }
    }
    sRedV[tid] = bvv; sRedI[tid] = bii; __syncthreads();
    for (int s = 128; s > 0; s >>= 1) {
        if (tid < s) {
            float v2 = sRedV[tid + s]; int i2 = sRedI[tid + s];
            if (v2 < sRedV[tid] || (v2 == sRedV[tid] && i2 < sRedI[tid])) {
                sRedV[tid] = v2; sRedI[tid] = i2;
            }
        }
        __syncthreads();
    }
    if (tid == 0) {
        int c = sRedI[0];
        float px = gx[c], py = gy[c];
        hx_g[b * HMAX + L] = px;
        hy_g[b * HMAX + L] = py;
        t_g[b * HMAX + L]  = t;
        out[64 + (step * 64 + b) * 2 + 0] = px;
        out[64 + (step * 64 + b) * 2 + 1] = py;
    }
}

// ---------------------------------------------------------------------------
// Kernel 4: _gmm_call on the grown 400-point history. Block = batch,
// wave per row group, two-pass logsumexp over the strict lower triangle.
// logp_i = LSE_{j<i}(a_j - 0.5*sq_ij/s^2) - LSE_{j<i}(a_j) + (-2 log s - LOG2PI)
// ---------------------------------------------------------------------------
__global__ __launch_bounds__(256) void gmm_kernel(
    const float* __restrict__ hx_g, const float* __restrict__ hy_g,
    const float* __restrict__ t_g,
    const float* __restrict__ beta_p, const float* __restrict__ log_sigma_p,
    float* __restrict__ out)
{
    __shared__ float sx[HMAX], sy[HMAX], st[HMAX];
    __shared__ float wsum[8];
    const int b = blockIdx.x, tid = threadIdx.x;
    const float beta = beta_p[0];
    const float ls = log_sigma_p[0];
    const float sigma = __expf(ls);
    const float inv2s2 = 0.5f / (sigma * sigma);
    const float lconst = -2.0f * ls - LOG2PIF;

    for (int j = tid; j < HMAX; j += 256) {
        sx[j] = hx_g[b * HMAX + j];
        sy[j] = hy_g[b * HMAX + j];
        st[j] = t_g[b * HMAX + j];
    }
    __syncthreads();

    const int lane = tid & 31, wv = tid >> 5;
    float local = 0.f;
    for (int i = 1 + wv; i < HMAX; i += 8) {
        const float ti = st[i], xi = sx[i], yi = sy[i];
        float m1 = -1e30f, m2 = -1e30f;
        for (int j = lane; j < i; j += 32) {
            float a = -beta * (ti - st[j]);
            float dx = xi - sx[j], dy = yi - sy[j];
            float bb = a - inv2s2 * (dx * dx + dy * dy);
            m1 = fmaxf(m1, a); m2 = fmaxf(m2, bb);
        }
        for (int o = 16; o > 0; o >>= 1) {
            m1 = fmaxf(m1, __shfl_xor(m1, o, 32));
            m2 = fmaxf(m2, __shfl_xor(m2, o, 32));
        }
        float s1 = 0.f, s2 = 0.f;
        for (int j = lane; j < i; j += 32) {
            float a = -beta * (ti - st[j]);
            float dx = xi - sx[j], dy = yi - sy[j];
            float bb = a - inv2s2 * (dx * dx + dy * dy);
            s1 += __expf(a - m1);
            s2 += __expf(bb - m2);
        }
        for (int o = 16; o > 0; o >>= 1) {
            s1 += __shfl_xor(s1, o, 32);
            s2 += __shfl_xor(s2, o, 32);
        }
        local += (m2 + __logf(s2)) - (m1 + __logf(s1));
    }
    if (lane == 0) wsum[wv] = local;
    __syncthreads();
    if (tid == 0) {
        float tot = 0.f;
        for (int w = 0; w < 8; ++w) tot += wsum[w];
        tot += 399.0f * lconst;                                    // per-row constant
        tot += -0.5f * (sx[0] * sx[0] + sy[0] * sy[0]) - LOG2PIF;  // logp0
        out[b] = tot;
    }
}

// ---------------------------------------------------------------------------
extern "C" void kernel_launch(void* const* d_in, const int* in_sizes, int n_in,
                              void* d_out, int out_size, void* d_ws, size_t ws_size,
                              hipStream_t stream) {
    const float* curr_time  = (const float*)d_in[0];
    const float* input_time = (const float*)d_in[1];
    const float* history    = (const float*)d_in[2];
    // d_in[3..5] (expected_data, aux_state_*) are dead in the reference.
    const float* beta       = (const float*)d_in[6];
    const float* log_sigma  = (const float*)d_in[7];
    float* out = (float*)d_out;

    float* ws = (float*)d_ws;
    float* gx = ws;                 // 640
    float* gy = gx + NCPAD;         // 640
    float* gn = gy + NCPAD;         // 640
    float* hx = gn + NCPAD;         // 64*400
    float* hy = hx + 64 * HMAX;     // 64*400
    float* tt = hy + 64 * HMAX;     // 64*400

    setup_grid_kernel<<<1, 64, 0, stream>>>(history, gx, gy, gn);
    init_hist_kernel<<<(64 * 384 + 255) / 256, 256, 0, stream>>>(history, input_time,
                                                                 hx, hy, tt);
    for (int i = 0; i < 16; ++i) {
        step_kernel<<<64, 256, 0, stream>>>(curr_time, beta, log_sigma,
                                            gx, gy, gn, hx, hy, tt, out, i);
    }
    gmm_kernel<<<64, 256, 0, stream>>>(hx, hy, tt, beta, log_sigma, out);
}